// QConv2d_3281355014838
// MI455X (gfx1250) — compile-verified
//
#include <hip/hip_runtime.h>
#include <math.h>

typedef __attribute__((ext_vector_type(2))) float v2f;
typedef __attribute__((ext_vector_type(8))) float v8f;

#define BATCH   512
#define HOUT    31
#define LPATCH  (HOUT*HOUT)          // 961
#define NPATCH  (BATCH*LPATCH)       // 492032
#define NTILES  (NPATCH/16)          // 30752 (exact)
#define NKER    4

__device__ __forceinline__ void cmul(float& ar, float& ai, float br, float bi) {
    float r = ar*br - ai*bi;
    float i = ar*bi + ai*br;
    ar = r; ai = i;
}

// ws layout: float W[kk][part(re=0,im=1)][kstep][vgpr][lane]  -> 4*2*4*2*32 = 2048 floats
// Element stored = B[k][n] = U[n][k] fragment value for that lane, where
//   n    = lane & 15                            (output amplitude / B column)
//   k    = 4*kstep + (lane&16 ? 2+vgpr : vgpr)  (input amplitude / B row)
__global__ void prep_bfrag(const float* __restrict__ w, float* __restrict__ W) {
    int idx  = blockIdx.x * 256 + threadIdx.x;   // 0..2047
    int lane = idx & 31;
    int v    = (idx >> 5) & 1;
    int step = (idx >> 6) & 3;
    int part = (idx >> 8) & 1;
    int kk   = (idx >> 9) & 3;
    int n    = lane & 15;
    int krow = 4*step + ((lane & 16) ? (2 + v) : v);
    // U[n][krow] = prod_q u3_q[bit_q(n)][bit_q(krow)], wire q = bit (3-q) (MSB first)
    float ar = 1.f, ai = 0.f;
    #pragma unroll
    for (int q = 0; q < 4; ++q) {
        const float* wq = w + (kk*4 + q)*3;
        float th = wq[0], ph = wq[1], om = wq[2];
        float s, c;
        __sincosf(0.5f*th, &s, &c);
        int jb = (n    >> (3 - q)) & 1;
        int ib = (krow >> (3 - q)) & 1;
        float br, bi;
        if (jb == 0 && ib == 0)      { br = c;  bi = 0.f; }
        else if (jb == 0 && ib == 1) { float so, co; __sincosf(om, &so, &co);       br = -co*s; bi = -so*s; }
        else if (jb == 1 && ib == 0) { float sp, cp; __sincosf(ph, &sp, &cp);       br =  cp*s; bi =  sp*s; }
        else                         { float sc, cc; __sincosf(ph + om, &sc, &cc);  br =  cc*c; bi =  sc*c; }
        cmul(ar, ai, br, bi);
    }
    W[idx] = part ? ai : ar;
}

__global__ __launch_bounds__(256) void qconv_wmma(
    const float* __restrict__ x, const float* __restrict__ W,
    float* __restrict__ out)
{
    const int lane = threadIdx.x & 31;
    const int tile = blockIdx.x * 8 + (threadIdx.x >> 5);
    const int half = (lane >> 4) & 1;      // 0: lanes 0-15, 1: lanes 16-31
    const int m16  = lane & 15;

    // ---- load + normalize this lane's patch (patch id = tile*16 + m16) ----
    int p  = tile*16 + m16;
    int b  = p / LPATCH;
    int l  = p - b*LPATCH;
    int ho = l / HOUT;
    int wo = l - ho*HOUT;
    const float* xb = x + ((size_t)b << 12) + (ho*2)*64 + (wo*2);
    float pa[16];
    #pragma unroll
    for (int kh = 0; kh < 4; ++kh) {
        const float* rp = xb + kh*64;
        float2 e0 = *reinterpret_cast<const float2*>(rp);
        float2 e1 = *reinterpret_cast<const float2*>(rp + 2);
        pa[kh*4+0] = e0.x; pa[kh*4+1] = e0.y;
        pa[kh*4+2] = e1.x; pa[kh*4+3] = e1.y;
    }
    float ss = 0.f;
    #pragma unroll
    for (int i = 0; i < 16; ++i) ss += pa[i]*pa[i];
    float inv1  = 1.f / (sqrtf(ss) + 1e-12f);   // amps = patch/(||patch||+eps)
    float ss2   = ss * inv1 * inv1;             // ||amps||^2
    float scale = inv1 / sqrtf(ss2);            // second normalize

    // CNOT-ring permutation: psi0[j] = amps[RINV[j]]
    const int RINV[16] = {0,13,3,14,6,11,5,8,12,1,15,2,10,7,9,4};
    float psi0[16];
    #pragma unroll
    for (int j = 0; j < 16; ++j) psi0[j] = pa[RINV[j]] * scale;

    // ---- A fragments (16x4 f32 layout): vgpr0={K0|K2}, vgpr1={K1|K3} ----
    v2f afrag[4];
    #pragma unroll
    for (int st = 0; st < 4; ++st) {
        afrag[st].x = psi0[4*st + (half ? 2 : 0)];
        afrag[st].y = psi0[4*st + (half ? 3 : 1)];
    }

    for (int kk = 0; kk < NKER; ++kk) {
        v8f dre = {};
        v8f dim_ = {};
        #pragma unroll
        for (int st = 0; st < 4; ++st) {
            int baseR = (kk*2 + 0)*256 + st*64 + lane;
            v2f bfr; bfr.x = W[baseR]; bfr.y = W[baseR + 32];
            dre = __builtin_amdgcn_wmma_f32_16x16x4_f32(
                false, afrag[st], false, bfr, (short)0, dre, false, false);
        }
        #pragma unroll
        for (int st = 0; st < 4; ++st) {
            int baseI = (kk*2 + 1)*256 + st*64 + lane;
            v2f bfi; bfi.x = W[baseI]; bfi.y = W[baseI + 32];
            dim_ = __builtin_amdgcn_wmma_f32_16x16x4_f32(
                false, afrag[st], false, bfi, (short)0, dim_, false, false);
        }

        // ---- observables per patch; D row M lives in one VGPR across 16 lanes ----
        #pragma unroll
        for (int r = 0; r < 8; ++r) {
            float tre = dre[r],  tim = dim_[r];
            float sre = __shfl_xor(tre, 8, 32);
            float sim = __shfl_xor(tim, 8, 32);
            float pA = tre*sre + tim*sim;          // Re(psi_j * conj(psi_{j+8})) pairs
            float pB = tim*sre - tre*sim;          // Im(...) (sign flips on upper 8)
            float mg = tre*tre + tim*tim;
            mg = (lane & 8) ? -mg : mg;            // +|psi_j|^2 (j<8), -(j>=8)
            pA += __shfl_xor(pA,1,32); pA += __shfl_xor(pA,2,32); pA += __shfl_xor(pA,4,32);
            pB += __shfl_xor(pB,1,32); pB += __shfl_xor(pB,2,32); pB += __shfl_xor(pB,4,32);
            mg += __shfl_xor(mg,1,32); mg += __shfl_xor(mg,2,32); mg += __shfl_xor(mg,4,32);
            float ez = mg + __shfl_xor(mg, 8, 32);
            if (m16 == 0) {                        // lane 0 -> patch r, lane 16 -> patch r+8
                int m  = r + (half ? 8 : 0);
                int pp = tile*16 + m;
                int bb = pp / LPATCH;
                int ll = pp - bb*LPATCH;
                float* o = out + ((size_t)bb*12 + kk*3)*LPATCH + ll;
                o[0]        =  2.f * pA;           // <X>
                o[LPATCH]   = -2.f * pB;           // <Y>
                o[2*LPATCH] =  ez;                 // <Z>
            }
        }
    }
}

extern "C" void kernel_launch(void* const* d_in, const int* in_sizes, int n_in,
                              void* d_out, int out_size, void* d_ws, size_t ws_size,
                              hipStream_t stream) {
    const float* x = (const float*)d_in[0];   // (512,1,64,64) f32
    const float* w = (const float*)d_in[1];   // (4,4,3) f32
    float* W   = (float*)d_ws;                // 2048 floats of B fragments
    float* out = (float*)d_out;               // (512,12,31,31) f32

    prep_bfrag<<<8, 256, 0, stream>>>(w, W);
    qconv_wmma<<<NTILES/8, 256, 0, stream>>>(x, W, out);
}